// GCN_45466523795657
// MI455X (gfx1250) — compile-verified
//
#include <hip/hip_runtime.h>

typedef float v2f __attribute__((ext_vector_type(2)));
typedef float v8f __attribute__((ext_vector_type(8)));

#define DFEAT 64
#define NEG_SLOPE 0.01f

__device__ __forceinline__ float leaky(float v) {
  return (v >= 0.0f) ? v : NEG_SLOPE * v;
}

// ---------------- degree / normalization prologue ----------------

__global__ void gcn_fill1(float* __restrict__ deg, int n) {
  int i = blockIdx.x * blockDim.x + threadIdx.x;
  if (i < n) deg[i] = 1.0f;  // self-loop included
}

__global__ void gcn_deg_accum(const int* __restrict__ dst, float* __restrict__ deg, int e) {
  int i = blockIdx.x * blockDim.x + threadIdx.x;
  if (i < e) atomicAdd(&deg[dst[i]], 1.0f);
}

__global__ void gcn_rsqrt(float* __restrict__ deg, int n) {
  int i = blockIdx.x * blockDim.x + threadIdx.x;
  if (i < n) deg[i] = rsqrtf(deg[i]);
}

// ---------------- fused GEMM + self-loop/bias epilogue ----------------
// H = act(X) @ W  (act = LeakyReLU for layers 2..4, identity for layer 1)
// also writes OUT = dinv[m]^2 * H + bias[n]  (scatter then accumulates on top)
//
// One wave -> one 16x16 tile. K=64 in 16 steps of V_WMMA_F32_16X16X4_F32.
// A (16x4 f32): lane(half,sub): v0 = X[m0+sub][k+2*half], v1 = [..+1]
// B (4x16 f32): lane(half,sub): v0 = W[k+2*half][n0+sub],  v1 = [..+1]
// C/D (16x16 f32): VGPR r, lane-half h -> row m0 + r + 8*h, col n0+sub
template <int APPLY_LEAKY>
__global__ void gcn_gemm_wmma(const float* __restrict__ X,
                              const float* __restrict__ W,
                              const float* __restrict__ bias,
                              const float* __restrict__ dinv,
                              float* __restrict__ H,
                              float* __restrict__ OUT, int nrows) {
  const int wave   = threadIdx.x >> 5;
  const int gw     = blockIdx.x * (blockDim.x >> 5) + wave;
  const int mtiles = (nrows + 15) >> 4;
  if (gw >= mtiles * 4) return;           // wave-uniform; EXEC all-1s inside
  const int mtile = gw >> 2;
  const int ntile = gw & 3;               // 4 consecutive waves share an A tile
  const int m0 = mtile << 4;
  const int n0 = ntile << 4;
  const int lane = threadIdx.x & 31;
  const int half = lane >> 4;
  const int sub  = lane & 15;

  if (m0 + 16 <= nrows) {
    v8f acc = {};
    const float* arow = X + (size_t)(m0 + sub) * DFEAT + 2 * half;
    const float* bcol = W + (size_t)(2 * half) * DFEAT + n0 + sub;
#pragma unroll
    for (int k = 0; k < DFEAT; k += 4) {
      v2f a, b;
      a.x = arow[k + 0];
      a.y = arow[k + 1];
      if (APPLY_LEAKY) { a.x = leaky(a.x); a.y = leaky(a.y); }
      b.x = bcol[(size_t)(k + 0) * DFEAT];
      b.y = bcol[(size_t)(k + 1) * DFEAT];
      acc = __builtin_amdgcn_wmma_f32_16x16x4_f32(
          /*neg_a=*/false, a, /*neg_b=*/false, b,
          /*c_mod=*/(short)0, acc, /*reuse_a=*/false, /*reuse_b=*/false);
    }
    const float bsub = bias[n0 + sub];
#pragma unroll
    for (int r = 0; r < 8; ++r) {
      const int m = m0 + r + 8 * half;
      const float di = dinv[m];
      const float hv = acc[r];
      H  [(size_t)m * DFEAT + n0 + sub] = hv;
      OUT[(size_t)m * DFEAT + n0 + sub] = di * di * hv + bsub;
    }
  } else {
    // scalar tail tile (wave-uniform branch; <16 rows remain)
    int mm = m0 + lane;
    if (mm < nrows) {
      const float di = dinv[mm];
      for (int nn = 0; nn < 16; ++nn) {
        float s = 0.0f;
        for (int k = 0; k < DFEAT; ++k) {
          float xv = X[(size_t)mm * DFEAT + k];
          if (APPLY_LEAKY) xv = leaky(xv);
          s += xv * W[(size_t)k * DFEAT + n0 + nn];
        }
        H  [(size_t)mm * DFEAT + n0 + nn] = s;
        OUT[(size_t)mm * DFEAT + n0 + nn] = di * di * s + bias[n0 + nn];
      }
    }
  }
}

// ---------------- edge scatter: out[dst] += dinv[src]*dinv[dst]*h[src] ----------------
// 2 edges per wave; 16 lanes per edge; each lane: one float4 gather + 4 f32 atomics.
// out and h both live in L2 (25.6 MB each << 192 MB), so this never leaves the chip.
__global__ void gcn_scatter(const float* __restrict__ h, const float* __restrict__ dinv,
                            const int* __restrict__ src, const int* __restrict__ dst,
                            float* __restrict__ out, int ne) {
  const int wv   = blockIdx.x * (blockDim.x >> 5) + (threadIdx.x >> 5);
  const int lane = threadIdx.x & 31;
  const int e    = wv * 2 + (lane >> 4);
  if (e >= ne) return;
  const int f = (lane & 15) * 4;
  const int s = src[e];
  const int d = dst[e];
  const float c = dinv[s] * dinv[d];
  const float4 v = *(const float4*)(h + (size_t)s * DFEAT + f);
  float* o = out + (size_t)d * DFEAT + f;
  atomicAdd(o + 0, c * v.x);
  atomicAdd(o + 1, c * v.y);
  atomicAdd(o + 2, c * v.z);
  atomicAdd(o + 3, c * v.w);
}

// ---------------- final LeakyReLU (only for layer 4's output) ----------------
__global__ void gcn_leaky(float* __restrict__ x, int n) {
  int t = blockIdx.x * blockDim.x + threadIdx.x;
  if (t < n) x[t] = leaky(x[t]);
}

extern "C" void kernel_launch(void* const* d_in, const int* in_sizes, int n_in,
                              void* d_out, int out_size, void* d_ws, size_t ws_size,
                              hipStream_t stream) {
  const float* x  = (const float*)d_in[0];
  const int*   ei = (const int*)d_in[1];
  const float* Wl[4] = {(const float*)d_in[2], (const float*)d_in[4],
                        (const float*)d_in[6], (const float*)d_in[8]};
  const float* bl[4] = {(const float*)d_in[3], (const float*)d_in[5],
                        (const float*)d_in[7], (const float*)d_in[9]};
  const int n = in_sizes[0] / DFEAT;
  const int e = in_sizes[1] / 2;
  const int* src = ei;
  const int* dst = ei + e;

  // workspace carve: dinv[n] | h[n*64] | t0[n*64] | t1[n*64]
  float* dinv = (float*)d_ws;
  float* h    = dinv + n;
  float* t0   = h  + (size_t)n * DFEAT;
  float* t1   = t0 + (size_t)n * DFEAT;
  float* outf = (float*)d_out;

  const int TPB = 256;

  // normalization coefficients (dinv = rsqrt(1 + in-degree))
  gcn_fill1    <<<(n + TPB - 1) / TPB, TPB, 0, stream>>>(dinv, n);
  gcn_deg_accum<<<(e + TPB - 1) / TPB, TPB, 0, stream>>>(dst, dinv, e);
  gcn_rsqrt    <<<(n + TPB - 1) / TPB, TPB, 0, stream>>>(dinv, n);

  const int mtiles        = (n + 15) / 16;
  const int gemm_blocks   = (mtiles * 4 + 7) / 8;     // 8 waves / 256-thread block
  const int scatter_waves = (e + 1) / 2;              // 2 edges per wave
  const int scatter_blks  = (scatter_waves + 7) / 8;

  const float* in = x;
  for (int l = 0; l < 4; ++l) {
    float* out = (l == 3) ? outf : ((l & 1) ? t1 : t0);
    if (l == 0)
      gcn_gemm_wmma<0><<<gemm_blocks, TPB, 0, stream>>>(in, Wl[l], bl[l], dinv, h, out, n);
    else
      gcn_gemm_wmma<1><<<gemm_blocks, TPB, 0, stream>>>(in, Wl[l], bl[l], dinv, h, out, n);
    gcn_scatter<<<scatter_blks, TPB, 0, stream>>>(h, dinv, src, dst, out, e);
    in = out;
  }
  gcn_leaky<<<(n * DFEAT + TPB - 1) / TPB, TPB, 0, stream>>>(outf, n * DFEAT);
}